// RhythmNetLoss_71201967833659
// MI455X (gfx1250) — compile-verified
//
#include <hip/hip_runtime.h>
#include <hip/hip_bf16.h>

typedef float v2f __attribute__((ext_vector_type(2)));
typedef float v8f __attribute__((ext_vector_type(8)));

#define NBLK   1024
#define NTHR   256
#define TOTAL  33554432LL              // 8192*4096
#define ITER   32                      // (TOTAL/4) / (NBLK*NTHR) float4 iters per thread
#define NX     24576                   // 6*4096 smooth slice
#define LAMBDA 150.0f

// -------- Stage 1: |a-b| partial sums, accumulated through V_WMMA_F32_16X16X4_F32 --------
// A (16x4 f32) carries 2 abs-diff values per lane (64 per wave per wmma), B = ones,
// C (v8f) accumulates row sums. Each lane's sum of its 8 acc regs is replicated 16x
// across the half-wave, so the block tree-sum is exactly 16 * true_sum -> scale by 1/16.
__global__ __launch_bounds__(NTHR) void l1_partial_kernel(const float* __restrict__ a,
                                                          const float* __restrict__ b,
                                                          float* __restrict__ partial) {
    const float4* __restrict__ a4 = reinterpret_cast<const float4*>(a);
    const float4* __restrict__ b4 = reinterpret_cast<const float4*>(b);

    long long idx = (long long)blockIdx.x * NTHR + threadIdx.x;   // in float4 units
    const long long stride = (long long)NBLK * NTHR;

    v8f acc = {};                     // 16x16 f32 C/D accumulator
    v2f ones; ones.x = 1.0f; ones.y = 1.0f;

#pragma unroll 4
    for (int it = 0; it < ITER; ++it, idx += stride) {
        float4 av = a4[idx];          // global_load_b128
        float4 bv = b4[idx];
        float d0 = __builtin_fabsf(av.x - bv.x);
        float d1 = __builtin_fabsf(av.y - bv.y);
        float d2 = __builtin_fabsf(av.z - bv.z);
        float d3 = __builtin_fabsf(av.w - bv.w);

        v2f a0; a0.x = d0; a0.y = d1;
        v2f a1; a1.x = d2; a1.y = d3;
        // D = A x ones + C : adds each lane-pair into per-row running sums on the XDL pipe
        acc = __builtin_amdgcn_wmma_f32_16x16x4_f32(false, a0, false, ones,
                                                    (short)0, acc, false, false);
        acc = __builtin_amdgcn_wmma_f32_16x16x4_f32(false, a1, false, ones,
                                                    (short)0, acc, false, false);
    }

    float s = acc[0] + acc[1] + acc[2] + acc[3] + acc[4] + acc[5] + acc[6] + acc[7];

    __shared__ float sdata[NTHR];
    sdata[threadIdx.x] = s;
    __syncthreads();
#pragma unroll
    for (int st = NTHR / 2; st > 0; st >>= 1) {
        if (threadIdx.x < st) sdata[threadIdx.x] += sdata[threadIdx.x + st];
        __syncthreads();
    }
    if (threadIdx.x == 0) partial[blockIdx.x] = sdata[0] * 0.0625f;  // exact /16
}

// -------- Stage 2: single-block finalize (deterministic) --------
__global__ __launch_bounds__(NTHR) void finalize_kernel(const float* __restrict__ partial,
                                                        const float* __restrict__ gru,
                                                        float* __restrict__ out) {
    __shared__ float sdata[NTHR];
    __shared__ float bcast[2];
    const int tid = threadIdx.x;

    // Phase 1: reduce the NBLK block partials -> l1 sum
    float s = 0.0f;
    for (int i = tid; i < NBLK; i += NTHR) s += partial[i];
    sdata[tid] = s;
    __syncthreads();
#pragma unroll
    for (int st = NTHR / 2; st > 0; st >>= 1) {
        if (tid < st) sdata[tid] += sdata[tid + st];
        __syncthreads();
    }
    if (tid == 0) bcast[0] = sdata[0];
    __syncthreads();

    // Phase 2: mean of x = gru[0 : 6*4096]
    const float4* __restrict__ x4 = reinterpret_cast<const float4*>(gru);
    const int n4 = NX / 4;            // 6144 float4
    float sx = 0.0f;
    for (int i = tid; i < n4; i += NTHR) {
        float4 v = x4[i];
        sx += v.x + v.y + v.z + v.w;
    }
    __syncthreads();                  // sdata free for reuse
    sdata[tid] = sx;
    __syncthreads();
#pragma unroll
    for (int st = NTHR / 2; st > 0; st >>= 1) {
        if (tid < st) sdata[tid] += sdata[tid + st];
        __syncthreads();
    }
    if (tid == 0) bcast[1] = sdata[0] / (float)NX;   // mean(x)
    __syncthreads();
    const float mean_x = bcast[1];

    // Phase 3: sum |x - mean(x)|
    float sa = 0.0f;
    for (int i = tid; i < n4; i += NTHR) {
        float4 v = x4[i];
        sa += __builtin_fabsf(v.x - mean_x) + __builtin_fabsf(v.y - mean_x)
            + __builtin_fabsf(v.z - mean_x) + __builtin_fabsf(v.w - mean_x);
    }
    __syncthreads();
    sdata[tid] = sa;
    __syncthreads();
#pragma unroll
    for (int st = NTHR / 2; st > 0; st >>= 1) {
        if (tid < st) sdata[tid] += sdata[tid + st];
        __syncthreads();
    }
    if (tid == 0) {
        const float l1     = bcast[0] / (float)TOTAL;
        const float smooth = sdata[0] / (float)NX;
        out[0] = l1 + LAMBDA * smooth;
    }
}

extern "C" void kernel_launch(void* const* d_in, const int* in_sizes, int n_in,
                              void* d_out, int out_size, void* d_ws, size_t ws_size,
                              hipStream_t stream) {
    (void)in_sizes; (void)n_in; (void)out_size; (void)ws_size;
    const float* gru    = (const float*)d_in[0];
    const float* target = (const float*)d_in[1];
    float* out     = (float*)d_out;
    float* partial = (float*)d_ws;    // NBLK floats of scratch, rewritten every call

    l1_partial_kernel<<<NBLK, NTHR, 0, stream>>>(gru, target, partial);
    finalize_kernel<<<1, NTHR, 0, stream>>>(partial, gru, out);
}